// Transformer_69956427317611
// MI455X (gfx1250) — compile-verified
//
#include <hip/hip_runtime.h>
#include <hip/hip_bf16.h>

// ---------------------------------------------------------------------------
// Graph-transformer block for MI455X (gfx1250, wave32, WMMA).
// All GEMMs use V_WMMA_F32_16X16X4_F32 (fp32-exact, matrix pipe).
// The huge (i,j) Y tensor never touches HBM: fused per-row-i in LDS.
// ---------------------------------------------------------------------------

typedef float v2f __attribute__((ext_vector_type(2)));
typedef float v8f __attribute__((ext_vector_type(8)));

constexpr int Bn = 2;
constexpr int Nn = 256;
constexpr int Dx = 128;
constexpr int De = 64;
constexpr float EPSf = 1e-5f;

// ---------------- workspace layout (floats) ----------------
constexpr size_t SZ_ROW   = (size_t)Bn * Nn * Dx;   // 65536
constexpr size_t OFS_YVEC = 0;                      // 2*10*128 = 2560
constexpr size_t OFS_Q    = 2560;
constexpr size_t OFS_K    = OFS_Q   + SZ_ROW;
constexpr size_t OFS_V    = OFS_K   + SZ_ROW;
constexpr size_t OFS_Z1A  = OFS_V   + SZ_ROW;
constexpr size_t OFS_Z2A  = OFS_Z1A + SZ_ROW;
constexpr size_t OFS_Z1L  = OFS_Z2A + SZ_ROW;
constexpr size_t OFS_Z2L  = OFS_Z1L + SZ_ROW;
constexpr size_t OFS_Z1T  = OFS_Z2L + SZ_ROW;
constexpr size_t OFS_Z2T  = OFS_Z1T + SZ_ROW;
constexpr size_t OFS_YA   = OFS_Z2T + SZ_ROW;
constexpr size_t OFS_YL   = OFS_YA  + SZ_ROW;
constexpr size_t OFS_YT   = OFS_YL  + SZ_ROW;
constexpr size_t OFS_WV   = OFS_YT  + SZ_ROW;
constexpr size_t OFS_EST  = OFS_WV  + SZ_ROW;       // 2*256*4*64 = 131072

// ---------------- d_out layout (floats) ----------------
constexpr size_t OUT_X = 0;
constexpr size_t OUT_E = OUT_X + (size_t)Bn*Nn*Dx;            // 65536
constexpr size_t OUT_A = OUT_E + (size_t)Bn*Nn*Nn*De;         // +8388608
constexpr size_t OUT_L = OUT_A + (size_t)Bn*Nn*De;
constexpr size_t OUT_T = OUT_L + (size_t)Bn*Nn*De;
constexpr size_t OUT_y = OUT_T + (size_t)Bn*Nn*De;

// ---------------- WMMA helpers ----------------
__device__ __forceinline__ v8f wmma_k4(v2f a, v2f b, v8f c) {
  // D = A(16x4) * B(4x16) + C(16x16), fp32
  return __builtin_amdgcn_wmma_f32_16x16x4_f32(false, a, false, b, (short)0, c,
                                               false, false);
}

// One 16x16 output tile: D[16, n0:n0+16] = A[16,K] * B[K,ldb] + bias
// A: row-major in LDS (lda), B: row-major (LDS or global, ldb), D: LDS (ldd).
__device__ __forceinline__ void gemm_tile(const float* As, int lda,
                                          const float* Bw, int ldb,
                                          float* Ds, int ldd,
                                          int n0, int K,
                                          const float* bias, bool relu) {
  const int lane = threadIdx.x & 31;
  const int lo = lane & 15;
  const int hi = lane >> 4;
  v8f c = {};
  const float* arow = As + lo * lda + 2 * hi;
  for (int k = 0; k < K; k += 4) {
    v2f a = *(const v2f*)(arow + k);                 // A: M=lo, K=k+2hi..+1
    v2f b;
    b.x = Bw[(size_t)(k + 2 * hi) * ldb + n0 + lo];  // B: K=k+2hi,   N=lo
    b.y = Bw[(size_t)(k + 2 * hi + 1) * ldb + n0 + lo];
    c = wmma_k4(a, b, c);
  }
  const float bb = bias ? bias[n0 + lo] : 0.f;
  for (int r = 0; r < 8; ++r) {
    float v = c[r] + bb;
    if (relu) v = fmaxf(v, 0.f);
    Ds[(r + 8 * hi) * ldd + n0 + lo] = v;            // D: M=r+8hi, N=lo
  }
}

// Row-wise LayerNorm of a 16 x D tile in LDS, in place. All threads call.
__device__ __forceinline__ void row_ln(float* H, int D, const float* g,
                                       const float* bb, float* mus, float* rss,
                                       int tid) {
  if (tid < 16) {
    const int r = tid;
    float mu = 0.f;
    for (int c = 0; c < D; ++c) mu += H[r * D + c];
    mu /= (float)D;
    float var = 0.f;
    for (int c = 0; c < D; ++c) { float v = H[r * D + c] - mu; var += v * v; }
    mus[r] = mu;
    rss[r] = rsqrtf(var / (float)D + EPSf);
  }
  __syncthreads();
  for (int x = tid; x < 16 * D; x += blockDim.x) {
    const int r = x / D, c = x % D;
    H[x] = (H[x] - mus[r]) * rss[r] * g[c] + bb[c];
  }
}

// ============================================================================
// K0: 10 y-conditioning vectors per batch:  yvec[b][v][0:128]
// v: 0=x_add 1=x_mul 2=e_add 3=e_mul 4=a_add 5=a_mul 6=l_add 7=l_mul 8=t_add 9=t_mul
// ============================================================================
struct YVecArgs {
  const float* y;
  const float* w[10];
  const float* bias[10];
  float* out;
};

__global__ __launch_bounds__(256) void k_yvec(YVecArgs a) {
  for (int idx = threadIdx.x; idx < Bn * 10 * 128; idx += blockDim.x) {
    const int d = idx & 127;
    const int v = (idx >> 7) % 10;
    const int b = idx / 1280;
    float acc = a.bias[v][d];
    const float* w = a.w[v];
    for (int k = 0; k < 64; ++k) acc += a.y[b * 64 + k] * w[k * 128 + d];
    a.out[idx] = acc;
  }
}

// ============================================================================
// K1: Out[512, 128] = A[512, K] @ W[K, 128] + bias, masked per row.
// One wave per 16x16 tile. grid = (32, 8), block = 32.
// ============================================================================
__global__ __launch_bounds__(32) void k_rowgemm(const float* __restrict__ A,
                                                const float* __restrict__ W,
                                                const float* __restrict__ bias,
                                                const float* __restrict__ nm,
                                                float* __restrict__ Out, int K) {
  const int m0 = blockIdx.x * 16, n0 = blockIdx.y * 16;
  const int lane = threadIdx.x & 31, lo = lane & 15, hi = lane >> 4;
  v8f c = {};
  const float* arow = A + (size_t)(m0 + lo) * K + 2 * hi;
  for (int k = 0; k < K; k += 4) {
    v2f a = *(const v2f*)(arow + k);
    v2f b;
    b.x = W[(size_t)(k + 2 * hi) * Dx + n0 + lo];
    b.y = W[(size_t)(k + 2 * hi + 1) * Dx + n0 + lo];
    c = wmma_k4(a, b, c);
  }
  const float bb = bias[n0 + lo];
  for (int r = 0; r < 8; ++r) {
    const int row = m0 + r + 8 * hi;
    Out[(size_t)row * Dx + n0 + lo] = (c[r] + bb) * nm[row];
  }
}

// ============================================================================
// K2: fused pairwise kernel. One workgroup per (b, i); 128 threads (4 waves).
// All E-track weights cached in LDS (224 KB). Per 16-j tile:
//   stage E -> WMMA e_add/e_mul -> elementwise Y + pairmod + online softmax
//   -> WMMA e_out -> LN -> WMMA FFN(64->256->64) -> LN -> final E out.
// ============================================================================
// LDS offsets (floats)
constexpr int SM_WEA = 0;        // 64x128
constexpr int SM_WEM = 8192;     // 64x128
constexpr int SM_WEO = 16384;    // 128x64
constexpr int SM_WF1 = 24576;    // 64x256
constexpr int SM_WF2 = 40960;    // 256x64
constexpr int SM_BEA = 57344;    // 128
constexpr int SM_BEM = 57472;    // 128
constexpr int SM_BEO = 57600;    // 64
constexpr int SM_BF1 = 57664;    // 256
constexpr int SM_BF2 = 57920;    // 64
constexpr int SM_YEA = 57984;    // 128
constexpr int SM_YEM = 58112;    // 128
constexpr int SM_G1  = 58240;    // 64
constexpr int SM_B1  = 58304;    // 64
constexpr int SM_G2  = 58368;    // 64
constexpr int SM_B2  = 58432;    // 64
constexpr int SM_ET  = 58496;    // 16x64
constexpr int SM_E1  = 59520;    // 16x128 (also reused for newE_pre)
constexpr int SM_E2  = 61568;    // 16x128
constexpr int SM_H   = 63616;    // 16x64
constexpr int SM_U   = 64640;    // 16x256
constexpr int SM_O   = 68736;    // 16x64
constexpr int SM_MU  = 69760;    // 16
constexpr int SM_RS  = 69776;    // 16
constexpr int SM_TOT = 69792;    // 279168 bytes
static_assert(SM_TOT * 4 <= 320 * 1024, "LDS overflow");

struct PairArgs {
  const float* E;
  const float* nm;
  const float* Q;  const float* Kp; const float* V;
  const float* Z1A; const float* Z2A;
  const float* Z1L; const float* Z2L;
  const float* Z1T; const float* Z2T;
  const float* yvec;
  const float* We_add; const float* be_add;
  const float* We_mul; const float* be_mul;
  const float* We_out; const float* be_out;
  const float* Wff1;   const float* bff1;
  const float* Wff2;   const float* bff2;
  const float* g1; const float* b1; const float* g2; const float* b2;
  float* outE;
  float* YA; float* YL; float* YT; float* WV; float* Estat;
};

__global__ __launch_bounds__(128) void k_pair(PairArgs p) {
  extern __shared__ float sm[];
  const int tid = threadIdx.x;
  const int b = blockIdx.x >> 8;
  const int i = blockIdx.x & 255;
  const int wv = tid >> 5;

  auto cpy = [&](int dst, const float* src, int n) {
    for (int x = tid; x < n; x += 128) sm[dst + x] = src[x];
  };
  cpy(SM_WEA, p.We_add, 8192);
  cpy(SM_WEM, p.We_mul, 8192);
  cpy(SM_WEO, p.We_out, 8192);
  cpy(SM_WF1, p.Wff1, 16384);
  cpy(SM_WF2, p.Wff2, 16384);
  cpy(SM_BEA, p.be_add, 128); cpy(SM_BEM, p.be_mul, 128);
  cpy(SM_BEO, p.be_out, 64);
  cpy(SM_BF1, p.bff1, 256);   cpy(SM_BF2, p.bff2, 64);
  cpy(SM_YEA, p.yvec + ((size_t)b * 10 + 2) * 128, 128);
  cpy(SM_YEM, p.yvec + ((size_t)b * 10 + 3) * 128, 128);
  cpy(SM_G1, p.g1, 64); cpy(SM_B1, p.b1, 64);
  cpy(SM_G2, p.g2, 64); cpy(SM_B2, p.b2, 64);
  __syncthreads();

  const float nmi = p.nm[b * Nn + i];
  const size_t rowi = (size_t)(b * Nn + i) * Dx + tid;
  const float Qd  = p.Q[rowi];
  const float z1a = p.Z1A[rowi];
  const float z1l = p.Z1L[rowi];
  const float z1t = p.Z1T[rowi];
  const float yea = sm[SM_YEA + tid];
  const float yem = sm[SM_YEM + tid];

  float accA = 0.f, accL = 0.f, accT = 0.f;
  float mval = -__builtin_inff(), ssum = 0.f, accV = 0.f;
  float esum = 0.f, esq = 0.f;
  float emin = __builtin_inff(), emax = -__builtin_inff();

  const size_t erow = (size_t)(b * Nn + i) * Nn;  // in units of De rows

  for (int jt = 0; jt < 16; ++jt) {
    const int j0 = jt * 16;
    // -- stage E tile (16 x 64)
    for (int x = tid; x < 16 * De; x += 128) {
      const int r = x >> 6, cc = x & 63;
      sm[SM_ET + x] = p.E[(erow + j0 + r) * De + cc];
    }
    if (jt < 15)  // prefetch next tile's first cacheline
      __builtin_prefetch(&p.E[(erow + j0 + 16) * De], 0, 1);
    __syncthreads();

    // -- E statistics for the double-PNA (threads 0..63 own one channel)
    if (tid < 64) {
      for (int r = 0; r < 16; ++r) {
        const float e = sm[SM_ET + r * 64 + tid];
        esum += e; esq += e * e;
        emin = fminf(emin, e); emax = fmaxf(emax, e);
      }
    }

    // -- E1 = E@We_add + b, E2 = E@We_mul + b   (16 tiles over 4 waves)
    for (int q = 0; q < 4; ++q) {
      const int combo = wv * 4 + q;
      const int which = combo >> 3;
      const int n0 = (combo & 7) * 16;
      gemm_tile(sm + SM_ET, 64, sm + (which ? SM_WEM : SM_WEA), 128,
                sm + (which ? SM_E2 : SM_E1), 128, n0, 64,
                sm + (which ? SM_BEM : SM_BEA), false);
    }
    __syncthreads();

    // -- elementwise: Y, pairmod accumulation, online softmax, newE_pre
    const int d = tid;
    for (int r = 0; r < 16; ++r) {
      const int j = j0 + r;
      const float nmj = p.nm[b * Nn + j];
      const float em = nmi * nmj;
      const size_t rowj = (size_t)(b * Nn + j) * Dx + d;
      const float kv = p.Kp[rowj];
      const float vv = p.V[rowj];
      const float Yv = Qd * kv * 0.25f;  // 1/sqrt(DF=16)
      const float e1 = sm[SM_E1 + r * 128 + d] * em;
      const float e2 = sm[SM_E2 + r * 128 + d] * em;
      const float Y = e1 + e2 * Yv + Yv;
      float z2, z3;
      z2 = p.Z2A[rowj]; z3 = z1a * z2; accA += z3 + z3 * Y + Y;
      z2 = p.Z2L[rowj]; z3 = z1l * z2; accL += z3 + z3 * Y + Y;
      z2 = p.Z2T[rowj]; z3 = z1t * z2; accT += z3 + z3 * Y + Y;
      if (nmj > 0.f) {  // key mask (f-wise softmax over j)
        const float mn = fmaxf(mval, Y);
        const float sc = __expf(mval - mn);
        const float ee = __expf(Y - mn);
        ssum = ssum * sc + ee;
        accV = accV * sc + ee * vv;
        mval = mn;
      }
      sm[SM_E1 + r * 128 + d] = yea + yem * Y + Y;  // newE_pre (reuse E1)
    }
    __syncthreads();

    // -- newE = newE_pre @ We_out + b  (4 tiles, one per wave)
    gemm_tile(sm + SM_E1, 128, sm + SM_WEO, 64, sm + SM_O, 64, wv * 16, 128,
              sm + SM_BEO, false);
    __syncthreads();

    // -- h = ln1_E(E + newE * Em)
    for (int x = tid; x < 16 * De; x += 128) {
      const int r = x >> 6;
      const float nmj = p.nm[b * Nn + j0 + r];
      sm[SM_H + x] = sm[SM_ET + x] + sm[SM_O + x] * (nmi * nmj);
    }
    __syncthreads();
    row_ln(sm + SM_H, 64, sm + SM_G1, sm + SM_B1, sm + SM_MU, sm + SM_RS, tid);
    __syncthreads();

    // -- u = relu(h @ Wff1 + b)   (16 tiles over 4 waves)
    for (int q = 0; q < 4; ++q)
      gemm_tile(sm + SM_H, 64, sm + SM_WF1, 256, sm + SM_U, 256,
                (wv * 4 + q) * 16, 64, sm + SM_BF1, true);
    __syncthreads();

    // -- o = u @ Wff2 + b
    gemm_tile(sm + SM_U, 256, sm + SM_WF2, 64, sm + SM_O, 64, wv * 16, 256,
              sm + SM_BF2, false);
    __syncthreads();

    // -- final = ln2_E(h + o) -> d_out E region
    if (tid < 16) {
      const int r = tid;
      float mu = 0.f;
      for (int cc = 0; cc < 64; ++cc)
        mu += sm[SM_H + r * 64 + cc] + sm[SM_O + r * 64 + cc];
      mu *= (1.f / 64.f);
      float var = 0.f;
      for (int cc = 0; cc < 64; ++cc) {
        const float v = sm[SM_H + r * 64 + cc] + sm[SM_O + r * 64 + cc] - mu;
        var += v * v;
      }
      sm[SM_MU + r] = mu;
      sm[SM_RS + r] = rsqrtf(var * (1.f / 64.f) + EPSf);
    }
    __syncthreads();
    for (int x = tid; x < 16 * De; x += 128) {
      const int r = x >> 6, cc = x & 63;
      const float v = sm[SM_H + x] + sm[SM_O + x];
      p.outE[(erow + j0 + r) * De + cc] =
          (v - sm[SM_MU + r]) * sm[SM_RS + r] * sm[SM_G2 + cc] + sm[SM_B2 + cc];
    }
    __syncthreads();
  }

  // -- epilogue: per-(b,i) reductions
  p.YA[rowi] = accA * (1.f / 256.f);
  p.YL[rowi] = accL * (1.f / 256.f);
  p.YT[rowi] = accT * (1.f / 256.f);
  p.WV[rowi] = (ssum > 0.f) ? (accV / ssum) : 0.f;
  if (tid < 64) {
    const size_t sbase = (size_t)(b * Nn + i) * 256;
    p.Estat[sbase + tid]       = esum;
    p.Estat[sbase + 64 + tid]  = esq;
    p.Estat[sbase + 128 + tid] = emin;
    p.Estat[sbase + 192 + tid] = emax;
  }
}

// ============================================================================
// K3: node-track outputs (X/A/L/T): out-proj + residual + LN + FFN + LN.
// grid = 32 blocks of 16 rows, 128 threads. Weights stream from L2.
// ============================================================================
constexpr int SMN_Z   = 0;       // 16x128
constexpr int SMN_H   = 2048;    // 16x128
constexpr int SMN_U   = 4096;    // 16x512
constexpr int SMN_O   = 12288;   // 16x128
constexpr int SMN_MU  = 14336;
constexpr int SMN_RS  = 14352;
constexpr int SMN_TOT = 14368;   // 57472 bytes

struct NodeArgs {
  const float* Xg; const float* Ag; const float* Lg; const float* Tg;
  const float* nm; const float* yvec;
  const float* WV; const float* YA; const float* YL; const float* YT;
  const float* Wxout; const float* bxout;
  const float* Waout; const float* baout;
  const float* Wdout; const float* bdout;
  const float* Wf1X; const float* bf1X; const float* Wf2X; const float* bf2X;
  const float* g1X; const float* b1X; const float* g2X; const float* b2X;
  const float* Wf1A; const float* bf1A; const float* Wf2A; const float* bf2A;
  const float* g1A; const float* b1A; const float* g2A; const float* b2A;
  const float* Wf1L; const float* bf1L; const float* Wf2L; const float* bf2L;
  const float* g1L; const float* b1L; const float* g2L; const float* b2L;
  const float* Wf1T; const float* bf1T; const float* Wf2T; const float* bf2T;
  const float* g1T; const float* b1T; const float* g2T; const float* b2T;
  float* outX; float* outA; float* outL; float* outT;
};

__device__ void track_pipeline(float* sm, int tid, int m0,
                               const float* src, const float* yadd,
                               const float* ymul, const float* Wout,
                               const float* bout, int Do, const float* orig,
                               const float* nm, const float* Wf1,
                               const float* bf1, int Dff, const float* Wf2,
                               const float* bf2, const float* g1,
                               const float* b1, const float* g2,
                               const float* b2, float* outp) {
  const int wv = tid >> 5;
  // z = yadd + (1 + ymul) * src
  for (int x = tid; x < 16 * 128; x += blockDim.x) {
    const int r = x >> 7, d = x & 127;
    const float s = src[(size_t)(m0 + r) * 128 + d];
    sm[SMN_Z + x] = yadd[d] + (1.f + ymul[d]) * s;
  }
  __syncthreads();
  // out-projection 128 -> Do
  const int ntile = Do / 16;
  for (int t = wv; t < ntile; t += 4)
    gemm_tile(sm + SMN_Z, 128, Wout, Do, sm + SMN_O, Do, t * 16, 128, bout,
              false);
  __syncthreads();
  // h = ln1(orig + proj * nm)
  for (int x = tid; x < 16 * Do; x += blockDim.x) {
    const int r = x / Do, d = x % Do;
    sm[SMN_H + x] = orig[(size_t)(m0 + r) * Do + d] + sm[SMN_O + x] * nm[m0 + r];
  }
  __syncthreads();
  row_ln(sm + SMN_H, Do, g1, b1, sm + SMN_MU, sm + SMN_RS, tid);
  __syncthreads();
  // u = relu(h @ Wf1 + b)
  for (int t = wv; t < Dff / 16; t += 4)
    gemm_tile(sm + SMN_H, Do, Wf1, Dff, sm + SMN_U, Dff, t * 16, Do, bf1, true);
  __syncthreads();
  // o = u @ Wf2 + b
  for (int t = wv; t < ntile; t += 4)
    gemm_tile(sm + SMN_U, Dff, Wf2, Do, sm + SMN_O, Do, t * 16, Dff, bf2,
              false);
  __syncthreads();
  // out = ln2(h + o)
  if (tid < 16) {
    const int r = tid;
    float mu = 0.f;
    for (int c = 0; c < Do; ++c) mu += sm[SMN_H + r * Do + c] + sm[SMN_O + r * Do + c];
    mu /= (float)Do;
    float var = 0.f;
    for (int c = 0; c < Do; ++c) {
      const float v = sm[SMN_H + r * Do + c] + sm[SMN_O + r * Do + c] - mu;
      var += v * v;
    }
    sm[SMN_MU + r] = mu;
    sm[SMN_RS + r] = rsqrtf(var / (float)Do + EPSf);
  }
  __syncthreads();
  for (int x = tid; x < 16 * Do; x += blockDim.x) {
    const int r = x / Do, d = x % Do;
    const float v = sm[SMN_H + x] + sm[SMN_O + x];
    outp[(size_t)(m0 + r) * Do + d] =
        (v - sm[SMN_MU + r]) * sm[SMN_RS + r] * g2[d] + b2[d];
  }
  __syncthreads();
}

__global__ __launch_bounds__(128) void k_node(NodeArgs a) {
  extern __shared__ float sm[];
  const int tid = threadIdx.x;
  const int m0 = blockIdx.x * 16;
  const int b = m0 / Nn;
  const float* yv = a.yvec + (size_t)b * 10 * 128;
  track_pipeline(sm, tid, m0, a.WV, yv + 0 * 128, yv + 1 * 128, a.Wxout,
                 a.bxout, 128, a.Xg, a.nm, a.Wf1X, a.bf1X, 512, a.Wf2X, a.bf2X,
                 a.g1X, a.b1X, a.g2X, a.b2X, a.outX);
  track_pipeline(sm, tid, m0, a.YA, yv + 4 * 128, yv + 5 * 128, a.Waout,
                 a.baout, 64, a.Ag, a.nm, a.Wf1A, a.bf1A, 256, a.Wf2A, a.bf2A,
                 a.g1A, a.b1A, a.g2A, a.b2A, a.outA);
  track_pipeline(sm, tid, m0, a.YL, yv + 6 * 128, yv + 7 * 128, a.Wdout,
                 a.bdout, 64, a.Lg, a.nm, a.Wf1L, a.bf1L, 256, a.Wf2L, a.bf2L,
                 a.g1L, a.b1L, a.g2L, a.b2L, a.outL);
  track_pipeline(sm, tid, m0, a.YT, yv + 8 * 128, yv + 9 * 128, a.Wdout,
                 a.bdout, 64, a.Tg, a.nm, a.Wf1T, a.bf1T, 256, a.Wf2T, a.bf2T,
                 a.g1T, a.b1T, a.g2T, a.b2T, a.outT);
}

// ============================================================================
// K4: y track: PNA(X,1) + PNA(E,2) + y MLP + FFN. One block, 256 threads.
// ============================================================================
struct YArgs {
  const float* X; const float* y; const float* Estat;
  const float* Wyy; const float* byy;
  const float* Wxy; const float* bxy;
  const float* Wey; const float* bey;
  const float* Wo1; const float* bo1; const float* Wo2; const float* bo2;
  const float* Wf1; const float* bf1; const float* Wf2; const float* bf2;
  const float* g1; const float* b1; const float* g2; const float* b2;
  float* outy;
};

__global__ __launch_bounds__(256) void k_y(YArgs a) {
  __shared__ float cx[2 * 512], ce[2 * 256], ny[2 * 64], h[2 * 64];
  __shared__ float u[2 * 256], o[2 * 64], r1[2 * 64];
  const int tid = threadIdx.x;
  // PNA over X (one reduction): [mean, min, max, std(ddof=1)]
  {
    const int b = tid >> 7, d = tid & 127;
    float s = 0.f, sq = 0.f, mn = __builtin_inff(), mx = -__builtin_inff();
    for (int i = 0; i < Nn; ++i) {
      const float v = a.X[((size_t)b * Nn + i) * Dx + d];
      s += v; sq += v * v; mn = fminf(mn, v); mx = fmaxf(mx, v);
    }
    const float m = s / Nn;
    const float sd = sqrtf(fmaxf(0.f, (sq - s * s / Nn) / (Nn - 1)));
    cx[b * 512 + d] = m;        cx[b * 512 + 128 + d] = mn;
    cx[b * 512 + 256 + d] = mx; cx[b * 512 + 384 + d] = sd;
  }
  // PNA over E (two reductions) from per-(b,i) stats
  if (tid < 128) {
    const int b = tid >> 6, d = tid & 63;
    float sm_ = 0.f, smin = __builtin_inff(), smax = -__builtin_inff();
    float ss = 0.f, ssq = 0.f;
    for (int i = 0; i < Nn; ++i) {
      const size_t base = (size_t)(b * Nn + i) * 256;
      const float su = a.Estat[base + d];
      const float q2 = a.Estat[base + 64 + d];
      const float s_i = sqrtf(fmaxf(0.f, (q2 - su * su / Nn) / (Nn - 1)));
      sm_ += su / Nn; ss += s_i; ssq += s_i * s_i;
      smin = fminf(smin, a.Estat[base + 128 + d]);
      smax = fmaxf(smax, a.Estat[base + 192 + d]);
    }
    const float m2 = sm_ / Nn;
    const float s2 = sqrtf(fmaxf(0.f, (ssq - ss * ss / Nn) / (Nn - 1)));
    ce[b * 256 + d] = m2;        ce[b * 256 + 64 + d] = smin;
    ce[b * 256 + 128 + d] = smax; ce[b * 256 + 192 + d] = s2;
  }
  __syncthreads();
  // new_y = y@Wyy + cx@Wxy + ce@Wey (+ biases)
  if (tid < 128) {
    const int b = tid >> 6, d = tid & 63;
    float acc = a.byy[d] + a.bxy[d] + a.bey[d];
    for (int k = 0; k < 64; ++k) acc += a.y[b * 64 + k] * a.Wyy[k * 64 + d];
    for (int k = 0; k < 512; ++k) acc += cx[b * 512 + k] * a.Wxy[k * 64 + d];
    for (int k = 0; k < 256; ++k) acc += ce[b * 256 + k] * a.Wey[k * 64 + d];
    ny[b * 64 + d] = acc;
  }
  __syncthreads();
  if (tid < 128) {
    const int b = tid >> 6, d = tid & 63;
    float acc = a.bo1[d];
    for (int k = 0; k < 64; ++k) acc += ny[b * 64 + k] * a.Wo1[k * 64 + d];
    r1[b * 64 + d] = fmaxf(acc, 0.f);
  }
  __syncthreads();
  if (tid < 128) {
    const int b = tid >> 6, d = tid & 63;
    float acc = a.bo2[d];
    for (int k = 0; k < 64; ++k) acc += r1[b * 64 + k] * a.Wo2[k * 64 + d];
    h[b * 64 + d] = a.y[b * 64 + d] + acc;   // residual before ln1
  }
  __syncthreads();
  if (tid < 2) {  // ln1 per batch
    const int b = tid;
    float mu = 0.f;
    for (int c = 0; c < 64; ++c) mu += h[b * 64 + c];
    mu /= 64.f;
    float var = 0.f;
    for (int c = 0; c < 64; ++c) { const float v = h[b * 64 + c] - mu; var += v * v; }
    const float rs = rsqrtf(var / 64.f + EPSf);
    for (int c = 0; c < 64; ++c)
      h[b * 64 + c] = (h[b * 64 + c] - mu) * rs * a.g1[c] + a.b1[c];
  }
  __syncthreads();
  for (int idx = tid; idx < 512; idx += blockDim.x) {
    const int b = idx >> 8, d = idx & 255;
    float acc = a.bf1[d];
    for (int k = 0; k < 64; ++k) acc += h[b * 64 + k] * a.Wf1[k * 256 + d];
    u[b * 256 + d] = fmaxf(acc, 0.f);
  }
  __syncthreads();
  if (tid < 128) {
    const int b = tid >> 6, d = tid & 63;
    float acc = a.bf2[d];
    for (int k = 0; k < 256; ++k) acc += u[b * 256 + k] * a.Wf2[k * 64 + d];
    o[b * 64 + d] = h[b * 64 + d] + acc;
  }
  __syncthreads();
  if (tid < 2) {  // ln2 per batch -> output
    const int b = tid;
    float mu = 0.f;
    for (int c = 0; c < 64; ++c) mu += o[b * 64 + c];
    mu /= 64.f;
    float var = 0.f;
    for (int c = 0; c < 64; ++c) { const float v = o[b * 64 + c] - mu; var += v * v; }
    const float rs = rsqrtf(var / 64.f + EPSf);
    for (int c = 0; c < 64; ++c)
      a.outy[b * 64 + c] = (o[b * 64 + c] - mu) * rs * a.g2[c] + a.b2[c];
  }
}

// ============================================================================
// host side
// ============================================================================
extern "C" void kernel_launch(void* const* d_in, const int* in_sizes, int n_in,
                              void* d_out, int out_size, void* d_ws,
                              size_t ws_size, hipStream_t stream) {
  const float* X  = (const float*)d_in[0];
  const float* E  = (const float*)d_in[1];
  const float* A  = (const float*)d_in[2];
  const float* L  = (const float*)d_in[3];
  const float* T  = (const float*)d_in[4];
  const float* y  = (const float*)d_in[5];
  const float* nm = (const float*)d_in[6];
  auto P = [&](int idx) { return (const float*)d_in[idx]; };
  float* ws  = (float*)d_ws;
  float* out = (float*)d_out;

  // K0: y-conditioning vectors
  YVecArgs ya;
  ya.y = y;
  ya.out = ws + OFS_YVEC;
  const int wi[10] = {31, 29, 35, 33, 39, 37, 43, 41, 47, 45};
  for (int v = 0; v < 10; ++v) { ya.w[v] = P(wi[v]); ya.bias[v] = P(wi[v] + 1); }
  k_yvec<<<1, 256, 0, stream>>>(ya);

  // K1: row projections (WMMA)
  dim3 g1(32, 8), b1(32);
  k_rowgemm<<<g1, b1, 0, stream>>>(X, P(7),  P(8),  nm, ws + OFS_Q,   128);
  k_rowgemm<<<g1, b1, 0, stream>>>(X, P(9),  P(10), nm, ws + OFS_K,   128);
  k_rowgemm<<<g1, b1, 0, stream>>>(X, P(11), P(12), nm, ws + OFS_V,   128);
  k_rowgemm<<<g1, b1, 0, stream>>>(A, P(17), P(18), nm, ws + OFS_Z1A, 64);
  k_rowgemm<<<g1, b1, 0, stream>>>(A, P(19), P(20), nm, ws + OFS_Z2A, 64);
  k_rowgemm<<<g1, b1, 0, stream>>>(L, P(21), P(22), nm, ws + OFS_Z1L, 64);
  k_rowgemm<<<g1, b1, 0, stream>>>(L, P(23), P(24), nm, ws + OFS_Z2L, 64);
  k_rowgemm<<<g1, b1, 0, stream>>>(T, P(25), P(26), nm, ws + OFS_Z1T, 64);
  k_rowgemm<<<g1, b1, 0, stream>>>(T, P(27), P(28), nm, ws + OFS_Z2T, 64);

  // K2: fused pairwise + E-track (WMMA, weights in LDS)
  PairArgs pa;
  pa.E = E; pa.nm = nm;
  pa.Q = ws + OFS_Q; pa.Kp = ws + OFS_K; pa.V = ws + OFS_V;
  pa.Z1A = ws + OFS_Z1A; pa.Z2A = ws + OFS_Z2A;
  pa.Z1L = ws + OFS_Z1L; pa.Z2L = ws + OFS_Z2L;
  pa.Z1T = ws + OFS_Z1T; pa.Z2T = ws + OFS_Z2T;
  pa.yvec = ws + OFS_YVEC;
  pa.We_add = P(13); pa.be_add = P(14);
  pa.We_mul = P(15); pa.be_mul = P(16);
  pa.We_out = P(57); pa.be_out = P(58);
  pa.Wff1 = P(75); pa.bff1 = P(76);
  pa.Wff2 = P(77); pa.bff2 = P(78);
  pa.g1 = P(79); pa.b1 = P(80); pa.g2 = P(81); pa.b2 = P(82);
  pa.outE = out + OUT_E;
  pa.YA = ws + OFS_YA; pa.YL = ws + OFS_YL; pa.YT = ws + OFS_YT;
  pa.WV = ws + OFS_WV; pa.Estat = ws + OFS_EST;
  k_pair<<<Bn * Nn, 128, SM_TOT * sizeof(float), stream>>>(pa);

  // K3: node tracks X/A/L/T
  NodeArgs na;
  na.Xg = X; na.Ag = A; na.Lg = L; na.Tg = T;
  na.nm = nm; na.yvec = ws + OFS_YVEC;
  na.WV = ws + OFS_WV; na.YA = ws + OFS_YA; na.YL = ws + OFS_YL;
  na.YT = ws + OFS_YT;
  na.Wxout = P(55); na.bxout = P(56);
  na.Waout = P(59); na.baout = P(60);
  na.Wdout = P(61); na.bdout = P(62);
  na.Wf1X = P(67); na.bf1X = P(68); na.Wf2X = P(69); na.bf2X = P(70);
  na.g1X = P(71); na.b1X = P(72); na.g2X = P(73); na.b2X = P(74);
  na.Wf1A = P(83); na.bf1A = P(84); na.Wf2A = P(85); na.bf2A = P(86);
  na.g1A = P(87); na.b1A = P(88); na.g2A = P(89); na.b2A = P(90);
  na.Wf1L = P(91); na.bf1L = P(92); na.Wf2L = P(93); na.bf2L = P(94);
  na.g1L = P(95); na.b1L = P(96); na.g2L = P(97); na.b2L = P(98);
  na.Wf1T = P(99); na.bf1T = P(100); na.Wf2T = P(101); na.bf2T = P(102);
  na.g1T = P(103); na.b1T = P(104); na.g2T = P(105); na.b2T = P(106);
  na.outX = out + OUT_X; na.outA = out + OUT_A;
  na.outL = out + OUT_L; na.outT = out + OUT_T;
  k_node<<<32, 128, SMN_TOT * sizeof(float), stream>>>(na);

  // K4: y track
  YArgs yy;
  yy.X = X; yy.y = y; yy.Estat = ws + OFS_EST;
  yy.Wyy = P(49); yy.byy = P(50);
  yy.Wxy = P(51); yy.bxy = P(52);
  yy.Wey = P(53); yy.bey = P(54);
  yy.Wo1 = P(63); yy.bo1 = P(64); yy.Wo2 = P(65); yy.bo2 = P(66);
  yy.Wf1 = P(107); yy.bf1 = P(108); yy.Wf2 = P(109); yy.bf2 = P(110);
  yy.g1 = P(111); yy.b1 = P(112); yy.g2 = P(113); yy.b2 = P(114);
  yy.outy = out + OUT_y;
  k_y<<<1, 256, 0, stream>>>(yy);
}